// MultinomialMaxPool2d_20512763805962
// MI455X (gfx1250) — compile-verified
//
#include <hip/hip_runtime.h>
#include <hip/hip_bf16.h>
#include <stdint.h>

// MultinomialMaxPool2d for MI455X (gfx1250).
// B=32, C=64, H=W=224, SPACING=2 -> ph=pw=112, 5 options per region (4 + null).
// Pure streaming kernel (~1.54 GB traffic, ~66 us at 23.3 TB/s HBM roofline).
// CDNA5 features: async global->LDS loads (ASYNCcnt) with NT temporal hints,
// s_wait_asynccnt split counter, wave32 layout, NT stores (192 MB L2 must not
// cache single-use lines). VALU trimmed via 2D grid (no div chains) and raw
// v_exp_f32/v_log_f32 base-2 scoring (argmax-equivalent transform).

#define TPB 256

static constexpr int Hc  = 224;
static constexpr int Wc  = 224;
static constexpr int PHc = 112;
static constexpr int PWc = 112;
static constexpr float EPSc = 1e-8f;
static constexpr float L2E  = 1.4426950408889634f;  // log2(e)
static constexpr float LN2  = 0.6931471805599453f;  // ln(2)

typedef float v2f __attribute__((ext_vector_type(2)));

__device__ __forceinline__ float rlog2(float x) { return __builtin_amdgcn_logf(x); }  // v_log_f32
__device__ __forceinline__ float rexp2(float x) { return __builtin_amdgcn_exp2f(x); } // v_exp_f32

__global__ __launch_bounds__(TPB)
void multinomial_maxpool2d_kernel(const float* __restrict__ act,
                                  const float* __restrict__ gum,
                                  float* __restrict__ out,
                                  int phpw,      // regions per plane (12544)
                                  int planes)    // B*C (2048)
{
    // Per-lane staging slot: 4 floats (2x2 activation tile) -> 16 B per thread.
    __shared__ float tile[TPB * 4];

    const int r    = blockIdx.x * TPB + threadIdx.x;  // region within plane
    const int bc   = blockIdx.y;                      // plane index (SALU-uniform)
    const bool live = (r < phpw);
    const int  rc   = live ? r : 0;

    // py = rc / 112 exactly for rc < 12544 via single 32-bit multiply.
    const int py = (rc * 9363) >> 20;
    const int px = rc - py * PWc;

    const size_t planeBase = (size_t)bc * (Hc * Wc);
    const int    inPlane   = (2 * py) * Wc + 2 * px;
    const float* p0 = act + planeBase + inPlane;       // row 2*py
    const float* p1 = p0 + Wc;                         // row 2*py+1

    // Low 32 bits of a generic pointer to __shared__ == LDS byte offset.
    const unsigned lbase = (unsigned)(size_t)(&tile[threadIdx.x * 4]);

    // CDNA5 async global->LDS copies (ASYNCcnt), non-temporal (single-use data).
    asm volatile("global_load_async_to_lds_b64 %0, %1, off th:TH_LOAD_NT"
                 :: "v"(lbase), "v"(p0) : "memory");
    asm volatile("global_load_async_to_lds_b64 %0, %1, off th:TH_LOAD_NT"
                 :: "v"(lbase + 8u), "v"(p1) : "memory");

    // Overlap: 5 gumbel uniforms (20 B/lane, contiguous across wave), NT loads.
    const int    t  = bc * phpw + rc;                  // flat region index
    const float* gp = gum + (size_t)t * 5;
    const float u0 = __builtin_nontemporal_load(gp + 0);
    const float u1 = __builtin_nontemporal_load(gp + 1);
    const float u2 = __builtin_nontemporal_load(gp + 2);
    const float u3 = __builtin_nontemporal_load(gp + 3);
    const float u4 = __builtin_nontemporal_load(gp + 4);

    // Gumbel inner term: n_j = -ln(u_j+eps) + eps = fma(-ln2, log2(u_j+eps), eps)
    const float n0 = fmaf(-LN2, rlog2(u0 + EPSc), EPSc);
    const float n1 = fmaf(-LN2, rlog2(u1 + EPSc), EPSc);
    const float n2 = fmaf(-LN2, rlog2(u2 + EPSc), EPSc);
    const float n3 = fmaf(-LN2, rlog2(u3 + EPSc), EPSc);
    const float n4 = fmaf(-LN2, rlog2(u4 + EPSc), EPSc);

    // Wait for the async tile copies, then read our own staged data (no
    // cross-wave sharing -> asynccnt alone is sufficient, no barrier).
    asm volatile("s_wait_asynccnt 0" ::: "memory");

    const int t4 = threadIdx.x * 4;
    const float a0 = tile[t4 + 0];
    const float a1 = tile[t4 + 1];
    const float a2 = tile[t4 + 2];
    const float a3 = tile[t4 + 3];

    // Stable softmax over {a0,a1,a2,a3,0}, exponentials in base 2.
    const float m  = fmaxf(fmaxf(fmaxf(a0, a1), fmaxf(a2, a3)), 0.0f);
    const float e0 = rexp2((a0 - m) * L2E);
    const float e1 = rexp2((a1 - m) * L2E);
    const float e2 = rexp2((a2 - m) * L2E);
    const float e3 = rexp2((a3 - m) * L2E);
    const float e4 = rexp2(-m * L2E);
    const float es = (e0 + e1 + e2 + e3 + e4) * EPSc;  // eps * sum

    // Argmax-equivalent score (uniform scale 1/ln2, common -log2(sum) dropped):
    //   score'_j = log2(e_j + eps*sum) - log2(n_j)
    const float s0 = rlog2(e0 + es) - rlog2(n0);
    const float s1 = rlog2(e1 + es) - rlog2(n1);
    const float s2 = rlog2(e2 + es) - rlog2(n2);
    const float s3 = rlog2(e3 + es) - rlog2(n3);
    const float s4 = rlog2(e4 + es) - rlog2(n4);

    int   w    = 0;
    float best = s0;
    if (s1 > best) { best = s1; w = 1; }
    if (s2 > best) { best = s2; w = 2; }
    if (s3 > best) { best = s3; w = 3; }
    if (s4 > best) { best = s4; w = 4; }

    if (live) {
        // sparse_detection: one-hot back in BCHW layout, two coalesced NT b64 stores
        v2f r0 = { w == 0 ? 1.0f : 0.0f, w == 1 ? 1.0f : 0.0f };
        v2f r1 = { w == 2 ? 1.0f : 0.0f, w == 3 ? 1.0f : 0.0f };
        float* sp = out + planeBase + inPlane;
        __builtin_nontemporal_store(r0, reinterpret_cast<v2f*>(sp));
        __builtin_nontemporal_store(r1, reinterpret_cast<v2f*>(sp + Wc));

        // pooled_map (1 unless null won) and winner index, concatenated flat.
        const size_t base = (size_t)planes * phpw * 4;   // = 4*R
        const size_t Rsz  = (size_t)planes * phpw;
        __builtin_nontemporal_store((w < 4) ? 1.0f : 0.0f, out + base + t);
        __builtin_nontemporal_store((float)w, out + base + Rsz + t);
    }
}

extern "C" void kernel_launch(void* const* d_in, const int* in_sizes, int n_in,
                              void* d_out, int out_size, void* d_ws, size_t ws_size,
                              hipStream_t stream) {
    const float* act = (const float*)d_in[0];   // [B,C,H,W] f32
    const float* gum = (const float*)d_in[1];   // [B,C,ph*pw,5] f32
    float* out = (float*)d_out;                 // sparse | pooled | winner

    const int phpw   = PHc * PWc;                       // 12544
    const int planes = in_sizes[0] / (Hc * Wc);         // 2048
    const int bx     = (phpw + TPB - 1) / TPB;          // 49 (exact)

    dim3 grid(bx, planes);
    multinomial_maxpool2d_kernel<<<grid, TPB, 0, stream>>>(act, gum, out,
                                                           phpw, planes);
}